// MeanAggregator_32925219291233
// MI455X (gfx1250) — compile-verified
//
#include <hip/hip_runtime.h>
#include <hip/hip_bf16.h>
#include <stdint.h>

// MeanAggregator: out[i,:] = (1/17) * sum over {i} ∪ 16 sampled neighbors of features[row,:]
// D = 128 f32 => one row = 512 B = exactly one global_load_async_to_lds_b128 per wave32.
//
// Roofline: 850K random 512B row gathers = 435 MB; feature table (51.2 MB) is L2-resident
// (192 MB L2) => L2-bandwidth-bound gather. No matrix op exists here, so no WMMA; the
// CDNA5 feature that fits is the ASYNCcnt-tracked global->LDS copy path, used as a
// double-buffered per-wave gather pipeline with indices software-pipelined 2 nodes ahead.

#define DFEAT 128
#define ROWB  512   // DFEAT * sizeof(float)

typedef float v4f __attribute__((ext_vector_type(4)));

// ---------------------------------------------------------------------------
// Async double-buffered aggregator, degree fixed at compile time (PER = 17).
// Block = 64 threads = 2 waves; LDS = 2 waves * 2 buffers * PER * 512 B = 34,816 B.
// ---------------------------------------------------------------------------
template <int PER>
__global__ __launch_bounds__(64) void agg_async_kernel(
    const float* __restrict__ feat,
    const int*   __restrict__ edst,
    float*       __restrict__ out,
    int nNodes, float scale)
{
    extern __shared__ __align__(16) char smem[];

    const int lane = threadIdx.x & 31;
    const int wIb  = threadIdx.x >> 5;                       // wave in block (0..1)
    const int gw   = (blockIdx.x * blockDim.x + threadIdx.x) >> 5;
    const int nw   = (gridDim.x * blockDim.x) >> 5;          // total waves

    int node = gw;
    if (node >= nNodes) return;

    const unsigned slotBytes = (unsigned)PER * ROWB;         // 8704 B per buffer
    const unsigned myBase    = (unsigned)(wIb * 2) * slotBytes;
    const unsigned long long gLane =
        (unsigned long long)(uintptr_t)feat + (unsigned long long)(lane * 16);

    // One NT vector load of a node's PER indices into lanes 0..PER-1 (read-once data).
    auto loadIdx = [&](int v) -> int {
        const int* dp = edst + (size_t)v * PER;
        return (lane < PER) ? __builtin_nontemporal_load(dp + lane) : 0;
    };

    // Issue PER async row-copies (global -> LDS, ASYNCcnt++) into buffer `buf`.
    auto issueRows = [&](int buf, int myIdx) {
        const unsigned lBase = myBase + (unsigned)buf * slotBytes + (unsigned)(lane * 16);
#pragma unroll
        for (int n = 0; n < PER; ++n) {
            int r = __shfl(myIdx, n, 32);                    // broadcast row index
            unsigned long long ga =
                gLane + (unsigned long long)(unsigned)r * (unsigned long long)ROWB;
            unsigned la = lBase + (unsigned)(n * ROWB);
            // one instruction copies a full 512B feature row into LDS
            asm volatile("global_load_async_to_lds_b128 %0, %1, off"
                         :: "v"(la), "v"(ga)
                         : "memory");
        }
    };

    // Reduce PER staged rows for node v from buffer `buf`, scale, NT-store.
    auto reduce = [&](int v, int buf) {
        const char* p = smem + myBase + (unsigned)buf * slotBytes + (unsigned)(lane * 16);
        float ax = 0.f, ay = 0.f, az = 0.f, aw = 0.f;
#pragma unroll
        for (int n = 0; n < PER; ++n) {
            const v4f t = *(const v4f*)(p + n * ROWB);       // ds_load_b128
            ax += t.x; ay += t.y; az += t.z; aw += t.w;
        }
        v4f r;
        r.x = ax * scale; r.y = ay * scale; r.z = az * scale; r.w = aw * scale;
        // write-once output: non-temporal so it doesn't evict the L2-resident table
        __builtin_nontemporal_store(r, (v4f*)(out + (size_t)v * DFEAT) + lane);
    };

    // ---- pipeline: rows prefetched 1 node ahead, indices 2 nodes ahead ----
    int n1 = node + nw;
    int n2 = n1 + nw;

    int idxCur = loadIdx(node);
    issueRows(0, idxCur);
    int idxN1 = (n1 < nNodes) ? loadIdx(n1) : 0;

    int buf = 0;
    while (n1 < nNodes) {
        issueRows(buf ^ 1, idxN1);                           // prefetch next node's rows
        int idxN2 = (n2 < nNodes) ? loadIdx(n2) : 0;         // start idx load (hidden below)
        // all but the newest PER async copies have landed => current node's rows ready
        asm volatile("s_wait_asynccnt %0" :: "n"(PER) : "memory");
        reduce(node, buf);                                   // hides idxN2 latency
        node = n1; n1 = n2; n2 += nw;
        idxN1 = idxN2;
        buf ^= 1;
    }
    asm volatile("s_wait_asynccnt 0" ::: "memory");
    reduce(node, buf);
}

// ---------------------------------------------------------------------------
// Generic fallback: direct gather to VGPRs, runtime degree.
// ---------------------------------------------------------------------------
__global__ __launch_bounds__(256) void agg_direct_kernel(
    const float* __restrict__ feat,
    const int*   __restrict__ edst,
    float*       __restrict__ out,
    int nNodes, int per, float scale)
{
    const int lane = threadIdx.x & 31;
    const int gw   = (blockIdx.x * blockDim.x + threadIdx.x) >> 5;
    const int nw   = (gridDim.x * blockDim.x) >> 5;

    for (int v = gw; v < nNodes; v += nw) {
        const int* dp = edst + (size_t)v * per;
        float ax = 0.f, ay = 0.f, az = 0.f, aw = 0.f;
        for (int n = 0; n < per; ++n) {
            int r = dp[n];
            const v4f t = *((const v4f*)(feat + (size_t)r * DFEAT) + lane);
            ax += t.x; ay += t.y; az += t.z; aw += t.w;
        }
        v4f o;
        o.x = ax * scale; o.y = ay * scale; o.z = az * scale; o.w = aw * scale;
        __builtin_nontemporal_store(o, (v4f*)(out + (size_t)v * DFEAT) + lane);
    }
}

// ---------------------------------------------------------------------------
// Inputs (setup_inputs order): 0=features [N_TOTAL*128 f32], 1=edge_seg [E i32],
// 2=edge_dst [E i32], 3=num_nodes [1 i32, device]. Output: [B*128] f32.
// ---------------------------------------------------------------------------
extern "C" void kernel_launch(void* const* d_in, const int* in_sizes, int n_in,
                              void* d_out, int out_size, void* d_ws, size_t ws_size,
                              hipStream_t stream) {
    (void)n_in; (void)d_ws; (void)ws_size;

    const float* feat = (const float*)d_in[0];
    const int*   edst = (const int*)d_in[2];
    float*       out  = (float*)d_out;

    const int Bn = out_size / DFEAT;                 // number of target nodes
    if (Bn <= 0) return;
    const int per = in_sizes[2] / Bn;                // degree incl. self-loop (17)
    const float scale = 1.0f / (float)per;

    if (per == 17) {
        constexpr int PER = 17;
        const int wavesPerBlock = 2;                 // 64 threads
        const int nodesPerWave  = 4;                 // multi-node pipeline per wave
        int blocks = (Bn + wavesPerBlock * nodesPerWave - 1) /
                     (wavesPerBlock * nodesPerWave);
        if (blocks < 1) blocks = 1;
        size_t shmem = (size_t)wavesPerBlock * 2 * PER * ROWB;   // 34,816 B
        agg_async_kernel<PER><<<blocks, wavesPerBlock * 32, shmem, stream>>>(
            feat, edst, out, Bn, scale);
    } else {
        int blocks = (Bn + 7) / 8;
        if (blocks < 1) blocks = 1;
        agg_direct_kernel<<<blocks, 256, 0, stream>>>(feat, edst, out, Bn, per, scale);
    }
}